// GraphCNLayer_39195871543809
// MI455X (gfx1250) — compile-verified
//
#include <hip/hip_runtime.h>

#define D 128

typedef __attribute__((ext_vector_type(2))) float v2f;
typedef __attribute__((ext_vector_type(8))) float v8f;

// ---------------- aggregation kernels ----------------

__global__ void zero_deg_kernel(float* __restrict__ deg, int n) {
    int i = blockIdx.x * blockDim.x + threadIdx.x;
    if (i < n) deg[i] = 0.0f;
}

__global__ void count_deg_kernel(const long long* __restrict__ src,
                                 const long long* __restrict__ dst,
                                 float* __restrict__ deg, int nE) {
    int e = blockIdx.x * blockDim.x + threadIdx.x;
    if (e < nE) {
        atomicAdd(&deg[(int)src[e]], 1.0f);
        atomicAdd(&deg[(int)dst[e]], 1.0f);
    }
}

__global__ void rsqrt_deg_kernel(float* __restrict__ deg, int n) {
    int i = blockIdx.x * blockDim.x + threadIdx.x;
    if (i < n) {
        float d = deg[i];
        deg[i] = (d > 0.0f) ? rsqrtf(d) : 0.0f;
    }
}

__global__ void copy_x_kernel(const float4* __restrict__ x,
                              float4* __restrict__ out, int n4) {
    int i = blockIdx.x * blockDim.x + threadIdx.x;
    if (i < n4) out[i] = x[i];
}

// one wave (32 lanes) per edge; each lane handles 4 channels (float4)
__global__ void edge_scatter_kernel(const long long* __restrict__ src,
                                    const long long* __restrict__ dst,
                                    const float* __restrict__ x,
                                    const float* __restrict__ dis,
                                    float* __restrict__ out, int nE) {
    int t = blockIdx.x * blockDim.x + threadIdx.x;
    int e = t >> 5;
    int lane = t & 31;
    if (e >= nE) return;
    int s = (int)src[e];
    int d = (int)dst[e];
    float norm = dis[s] * dis[d];                 // wave-uniform
    int c = lane * 4;
    const float4 xs = *(const float4*)(x + (size_t)s * D + c);
    const float4 xd = *(const float4*)(x + (size_t)d * D + c);
    float* od = out + (size_t)d * D + c;
    float* os = out + (size_t)s * D + c;
    atomicAdd(od + 0, norm * xs.x);
    atomicAdd(od + 1, norm * xs.y);
    atomicAdd(od + 2, norm * xs.z);
    atomicAdd(od + 3, norm * xs.w);
    atomicAdd(os + 0, norm * xd.x);
    atomicAdd(os + 1, norm * xd.y);
    atomicAdd(os + 2, norm * xd.z);
    atomicAdd(os + 3, norm * xd.w);
}

// ---------------- WMMA GEMM + bias + relu (in place on `out`) ----------------
// Each workgroup: 128 nodes x 128 out-channels. 8 waves; wave w owns rows
// [blk*128 + w*16, +16) and sweeps all 8 column tiles with 8 f32 accumulators.
// D = A(16x4 f32) x B(4x16 f32) + C, K looped 128/4 = 32 times.
//
// OOB rows are handled by CLAMPING the A-row address (not predication):
// garbage in A row m only corrupts D row m, and clamped lanes' output rows
// are >= nNodes and never stored. This keeps the inner loop free of
// exec-mask save/restore around the global A load.

__global__ void __launch_bounds__(256)
gemm_bias_relu_kernel(const float* __restrict__ W,     // [D_OUT][D_IN] row-major
                      const float* __restrict__ bias,  // [D_OUT]
                      float* __restrict__ out,         // [N][D] in/out
                      int nNodes) {
    __shared__ float ldsW[D * D];  // 64 KB, W[o][k]

    for (int i = threadIdx.x; i < (D * D) / 4; i += blockDim.x) {
        ((float4*)ldsW)[i] = ((const float4*)W)[i];
    }
    __syncthreads();

    const int wave = threadIdx.x >> 5;
    const int lane = threadIdx.x & 31;
    const int rowBase = blockIdx.x * 128 + wave * 16;
    const int mn  = lane & 15;          // A row (M) / B col (N), per fragment layout
    const int k2  = (lane >> 4) * 2;    // K-pair: lanes 0-15 -> K 0,1; 16-31 -> K 2,3

    v8f acc[8] = {};

    int node = rowBase + mn;
    if (node > nNodes - 1) node = nNodes - 1;   // clamp, see note above
    const float* aptr = out + (size_t)node * D + k2;

    for (int kb = 0; kb < D; kb += 4) {
        const v2f a = *(const v2f*)(aptr + kb);
#pragma unroll
        for (int t = 0; t < 8; ++t) {
            // B[k][o] = W[o][k]; o = t*16 + mn, contiguous K pair in LDS
            const v2f b = *(const v2f*)(&ldsW[(t * 16 + mn) * D + kb + k2]);
            acc[t] = __builtin_amdgcn_wmma_f32_16x16x4_f32(
                false, a, false, b, (short)0, acc[t], false, false);
        }
    }

    // C/D layout: VGPR i -> M = i + (lane>=16 ? 8 : 0), N = lane&15
    const int rowHalf = (lane >> 4) * 8;
#pragma unroll
    for (int t = 0; t < 8; ++t) {
        const int col = t * 16 + mn;
        const float bv = bias[col];
#pragma unroll
        for (int i = 0; i < 8; ++i) {
            int row = rowBase + rowHalf + i;
            if (row < nNodes) {
                float v = acc[t][i] + bv;
                out[(size_t)row * D + col] = (v > 0.0f) ? v : 0.0f;
            }
        }
    }
}

// ---------------- launcher ----------------

extern "C" void kernel_launch(void* const* d_in, const int* in_sizes, int n_in,
                              void* d_out, int out_size, void* d_ws, size_t ws_size,
                              hipStream_t stream) {
    const float*     x    = (const float*)d_in[0];
    const long long* ei   = (const long long*)d_in[1];   // int64 [2, E]
    const float*     W    = (const float*)d_in[2];
    const float*     bias = (const float*)d_in[3];
    float*           out  = (float*)d_out;

    const int nNodes = in_sizes[0] / D;
    const int nE     = in_sizes[1] / 2;
    const long long* src = ei;
    const long long* dst = ei + nE;

    float* deg = (float*)d_ws;   // N floats (~400 KB); becomes rsqrt(deg) in place

    zero_deg_kernel<<<(nNodes + 255) / 256, 256, 0, stream>>>(deg, nNodes);
    count_deg_kernel<<<(nE + 255) / 256, 256, 0, stream>>>(src, dst, deg, nE);
    rsqrt_deg_kernel<<<(nNodes + 255) / 256, 256, 0, stream>>>(deg, nNodes);

    const int n4 = nNodes * D / 4;
    copy_x_kernel<<<(n4 + 255) / 256, 256, 0, stream>>>((const float4*)x,
                                                        (float4*)out, n4);

    const int scatterThreads = nE * 32;   // one wave per edge
    edge_scatter_kernel<<<(scatterThreads + 255) / 256, 256, 0, stream>>>(
        src, dst, x, deg, out, nE);

    gemm_bias_relu_kernel<<<(nNodes + 127) / 128, 256, 0, stream>>>(
        W, bias, out, nNodes);
}